// DMLoss_61942018343083
// MI455X (gfx1250) — compile-verified
//
#include <hip/hip_runtime.h>
#include <hip/hip_bf16.h>

typedef __attribute__((ext_vector_type(2))) float v2f;
typedef __attribute__((ext_vector_type(8))) float v8f;

#define BATCH 32
#define NP 512
#define NG 512
#define TIME_F 10
#define NGT (NG * TIME_F)   // 5120

// ---------------------------------------------------------------------------
// K0: gt_interp[b, n*T + t] = gt[b,n]*(t/T) + gt[b,n-1]*(1 - t/T)  (roll by 1)
// ---------------------------------------------------------------------------
__global__ void interp_kernel(const float* __restrict__ gt,
                              float* __restrict__ gi) {
    int idx = blockIdx.x * blockDim.x + threadIdx.x;
    int total = BATCH * NGT;
    if (idx >= total) return;
    int b = idx / NGT;
    int j = idx - b * NGT;
    int n = j / TIME_F;
    int t = j - n * TIME_F;
    int nm1 = (n == 0) ? (NG - 1) : (n - 1);
    float s = (float)t / (float)TIME_F;
    const float* g = gt + (size_t)b * NG * 2;
    float x = g[2 * n]     * s + g[2 * nm1]     * (1.0f - s);
    float y = g[2 * n + 1] * s + g[2 * nm1 + 1] * (1.0f - s);
    gi[(size_t)idx * 2]     = x;
    gi[(size_t)idx * 2 + 1] = y;
}

// ---------------------------------------------------------------------------
// K1: for each column point c (16 per block), argmin over all row points r of
//     ||row_r - col_c||^2, via D = A x B with V_WMMA_F32_16X16X4_F32:
//       A[j] = (x_j, y_j, x_j^2+y_j^2, 0)   (rows)
//       B[i] = (-2x_i, -2y_i, 1, 0)^T       (cols)
//     D[j,i] = ||row_j||^2 - 2 row_j . col_i  (== dist^2 minus col-const)
// A-matrix f32 16x4 layout: v[0]={K0|K2}, v[1]={K1|K3} across lane halves.
// B-matrix f32 4x16 layout (row-striped): v[0]={K0 row | K2 row}, v[1]={K1|K3}.
// C/D layout: lane holds column N=lane&15; VGPR r holds rows r and r+8.
// One wave per block; EXEC is all-ones around the WMMA.
// ---------------------------------------------------------------------------
__global__ __launch_bounds__(32)
void argmin_rows_wmma(const float* __restrict__ rowPts,  // [B, Nrow, 2]
                      const float* __restrict__ colPts,  // [B, Ncol, 2]
                      int Nrow, int Ncol,
                      int* __restrict__ outIdx) {        // [B, Ncol]
    const int b    = blockIdx.y;
    const int c0   = blockIdx.x * 16;
    const int lane = threadIdx.x;
    const int l15  = lane & 15;
    const bool hi  = lane >= 16;
    const int halfOff = hi ? 8 : 0;

    const float* rp = rowPts + (size_t)b * Nrow * 2;
    const float* cp = colPts + (size_t)b * Ncol * 2;

    // Build B matrix once for this column tile.
    float cx = cp[2 * (c0 + l15)];
    float cy = cp[2 * (c0 + l15) + 1];
    v2f Bm;
    Bm.x = hi ? 1.0f : (-2.0f * cx);   // K=2 row : K=0 row
    Bm.y = hi ? 0.0f : (-2.0f * cy);   // K=3 row : K=1 row

    float minv = 3.4e38f;
    int   minj = 0;

    for (int j0 = 0; j0 < Nrow; j0 += 16) {
        int p = j0 + l15;
        float x = rp[2 * p];
        float y = rp[2 * p + 1];
        v2f Am;
        Am.x = hi ? (x * x + y * y) : x;   // K=2 col : K=0 col
        Am.y = hi ? 0.0f : y;              // K=3 col : K=1 col

        v8f C = {};
        v8f D = __builtin_amdgcn_wmma_f32_16x16x4_f32(
            /*neg_a=*/false, Am, /*neg_b=*/false, Bm,
            /*c_mod=*/(short)0, C, /*reuse_a=*/false, /*reuse_b=*/false);

#pragma unroll
        for (int r = 0; r < 8; ++r) {
            float dv = D[r];
            int row = j0 + r + halfOff;
            if (dv < minv) { minv = dv; minj = row; }
        }
    }

    // Combine the two lane halves (rows 0-7 vs 8-15 of each tile).
    float ov = __shfl_xor(minv, 16, 32);
    int   oj = __shfl_xor(minj, 16, 32);
    if (ov < minv || (ov == minv && oj < minj)) { minv = ov; minj = oj; }

    if (!hi) outIdx[(size_t)b * Ncol + c0 + l15] = minj;
}

// ---------------------------------------------------------------------------
// K2: per-batch partial sums:
//   s0 = sum_i |pred_polys_[b,i] - gt_interp[b, idxGt[b,i]]|_1
//   s1 = sum_j mask[b,j] * |pred_polys_[b, idxPred[b,j]] - gt[b,j]|_1
//   s2 = sum_j mask[b,j]
// ---------------------------------------------------------------------------
__global__ __launch_bounds__(256)
void partials_kernel(const float* __restrict__ pp,    // pred_polys_ [B,NP,2]
                     const float* __restrict__ gi,    // gt_interp   [B,NGT,2]
                     const float* __restrict__ gt,    // gt_polys    [B,NG,2]
                     const float* __restrict__ mask,  // [B,NG]
                     const int* __restrict__ idxGt,   // [B,NP]
                     const int* __restrict__ idxPred, // [B,NG]
                     float* __restrict__ partials) {  // [B,3]
    int b = blockIdx.x;
    int tid = threadIdx.x;
    const float* ppb = pp + (size_t)b * NP * 2;
    const float* gib = gi + (size_t)b * NGT * 2;
    const float* gtb = gt + (size_t)b * NG * 2;
    const float* mb  = mask + (size_t)b * NG;

    float s0 = 0.0f, s1 = 0.0f, s2 = 0.0f;
    for (int i = tid; i < NP; i += 256) {
        int j = idxGt[b * NP + i];
        s0 += fabsf(ppb[2 * i]     - gib[2 * j]) +
              fabsf(ppb[2 * i + 1] - gib[2 * j + 1]);
    }
    for (int j = tid; j < NG; j += 256) {
        int i = idxPred[b * NG + j];
        float m = mb[j];
        s1 += m * (fabsf(ppb[2 * i]     - gtb[2 * j]) +
                   fabsf(ppb[2 * i + 1] - gtb[2 * j + 1]));
        s2 += m;
    }

    __shared__ float sh0[256], sh1[256], sh2[256];
    sh0[tid] = s0; sh1[tid] = s1; sh2[tid] = s2;
    __syncthreads();
    for (int off = 128; off > 0; off >>= 1) {
        if (tid < off) {
            sh0[tid] += sh0[tid + off];
            sh1[tid] += sh1[tid + off];
            sh2[tid] += sh2[tid + off];
        }
        __syncthreads();
    }
    if (tid == 0) {
        partials[b * 3 + 0] = sh0[0];
        partials[b * 3 + 1] = sh1[0];
        partials[b * 3 + 2] = sh2[0];
    }
}

// ---------------------------------------------------------------------------
// K3: final scalar (one wave): out = (s1/(2*s2+1) + s0/(B*NP*2)) / 2
// ---------------------------------------------------------------------------
__global__ __launch_bounds__(32)
void final_kernel(const float* __restrict__ partials, float* __restrict__ out) {
    int t = threadIdx.x;
    float s0 = 0.0f, s1 = 0.0f, s2 = 0.0f;
    if (t < BATCH) {
        s0 = partials[t * 3 + 0];
        s1 = partials[t * 3 + 1];
        s2 = partials[t * 3 + 2];
    }
    for (int off = 16; off > 0; off >>= 1) {
        s0 += __shfl_down(s0, off, 32);
        s1 += __shfl_down(s1, off, 32);
        s2 += __shfl_down(s2, off, 32);
    }
    if (t == 0) {
        float pred2gt = s0 / (float)(BATCH * NP * 2);
        float loss = s1 / (2.0f * s2 + 1.0f) + pred2gt;
        out[0] = loss * 0.5f;
    }
}

// ---------------------------------------------------------------------------
extern "C" void kernel_launch(void* const* d_in, const int* in_sizes, int n_in,
                              void* d_out, int out_size, void* d_ws, size_t ws_size,
                              hipStream_t stream) {
    const float* ini_pred = (const float*)d_in[0]; // [B,NP,2]
    const float* pred_pp  = (const float*)d_in[1]; // [B,NP,2]
    const float* gt_polys = (const float*)d_in[2]; // [B,NG,2]
    const float* kpMask   = (const float*)d_in[3]; // [B,NG]
    float* out = (float*)d_out;

    // Workspace layout
    char* ws = (char*)d_ws;
    float* gi       = (float*)ws;                       // B*NGT*2 floats
    ws += (size_t)BATCH * NGT * 2 * sizeof(float);
    int*   idxGt    = (int*)ws;                         // B*NP ints
    ws += (size_t)BATCH * NP * sizeof(int);
    int*   idxPred  = (int*)ws;                         // B*NG ints
    ws += (size_t)BATCH * NG * sizeof(int);
    float* partials = (float*)ws;                       // B*3 floats

    // K0: interpolate gt polygon
    {
        int total = BATCH * NGT;
        interp_kernel<<<(total + 255) / 256, 256, 0, stream>>>(gt_polys, gi);
    }

    // K1a: pred -> nearest interpolated gt (rows = gt_interp, cols = ini_pred)
    {
        dim3 grid(NP / 16, BATCH);
        argmin_rows_wmma<<<grid, 32, 0, stream>>>(gi, ini_pred, NGT, NP, idxGt);
    }

    // K1b: gt -> nearest pred (rows = ini_pred, cols = gt_polys)
    {
        dim3 grid(NG / 16, BATCH);
        argmin_rows_wmma<<<grid, 32, 0, stream>>>(ini_pred, gt_polys, NP, NG, idxPred);
    }

    // K2: per-batch partial sums
    partials_kernel<<<BATCH, 256, 0, stream>>>(pred_pp, gi, gt_polys, kpMask,
                                               idxGt, idxPred, partials);

    // K3: final scalar
    final_kernel<<<1, 32, 0, stream>>>(partials, out);
}